// Add_Gryo_LEM_32323923870226
// MI455X (gfx1250) — compile-verified
//
#include <hip/hip_runtime.h>
#include <math.h>

// ---------------------------------------------------------------------------
// MI455X (gfx1250) batched SPD log-Euclidean update:
//   out[b] = expm( logm(x[b]) + 0.5*(bias + bias^T) )
// All matrix functions are 32x32 f32 GEMMs built from V_WMMA_F32_16X16X4_F32
// (wave32, 2x2 tiles of 16x16, 8 K-chunks of 4). One wave per matrix;
// matrices live in LDS, addressed as integer offsets into the extern-shared
// array so every access lowers to ds_load/ds_store (not flat).
// ---------------------------------------------------------------------------

#define STRIDE 33                 // 32 + 1 pad: kills LDS bank conflicts on column walks
#define MSZ (32 * STRIDE)         // floats per LDS matrix buffer
#define WAVES_PER_BLOCK 8
#define BUFS_PER_WAVE 4
#define LDS_BYTES ((size_t)WAVES_PER_BLOCK * BUFS_PER_WAVE * MSZ * sizeof(float)) // 135168

typedef float v2f __attribute__((ext_vector_type(2)));
typedef float v8f __attribute__((ext_vector_type(8)));

extern __shared__ float smem[];   // all helpers index this directly -> addrspace(3)

// Intra-wave LDS fence: waves own disjoint LDS regions, so no block barrier is
// needed -- only ordering of this wave's ds ops + a compiler memory fence.
// Uses the CDNA5 split dependency counter (s_wait_dscnt).
__device__ __forceinline__ void wave_sync() {
  __builtin_amdgcn_wave_barrier();
  asm volatile("s_wait_dscnt 0" ::: "memory");
  __builtin_amdgcn_wave_barrier();
}

// ---------------------------------------------------------------------------
// smem[cOff..] = smem[aOff..] * smem[bOff..] for 32x32 f32 (row pitch STRIDE).
// WMMA f32 16x16x4 operand layouts (ISA 7.12.2):
//   A 16x4 : lane l, vgpr v -> M = l%16, K = (l/16)*2 + v   (a.x/a.y contiguous
//            in a row -> single ds_load_b64)
//   B 4x16 : lane l, vgpr v -> K = (l/16)*2 + v, N = l%16
//   C 16x16: lane l, vgpr r -> M = (l/16)*8 + r, N = l%16
// Force-inlined so LDS address-space inference always succeeds.
// ---------------------------------------------------------------------------
__device__ __forceinline__ void mm32(int cOff, int aOff, int bOff, int lane) {
  const int m    = lane & 15;
  const int koff = (lane >> 4) << 1;
  const int half = lane >> 4;
  #pragma unroll
  for (int i = 0; i < 2; ++i) {
    #pragma unroll
    for (int j = 0; j < 2; ++j) {
      v8f acc = {0.f, 0.f, 0.f, 0.f, 0.f, 0.f, 0.f, 0.f};
      #pragma unroll
      for (int kk = 0; kk < 8; ++kk) {
        const int kb = kk * 4 + koff;
        v2f a, b;
        a.x = smem[aOff + (i * 16 + m) * STRIDE + kb];
        a.y = smem[aOff + (i * 16 + m) * STRIDE + kb + 1];
        b.x = smem[bOff + (kb)     * STRIDE + j * 16 + m];
        b.y = smem[bOff + (kb + 1) * STRIDE + j * 16 + m];
        // (neg_a, A, neg_b, B, c_mod, C, reuse_a, reuse_b)
        acc = __builtin_amdgcn_wmma_f32_16x16x4_f32(false, a, false, b,
                                                    (short)0, acc, false, false);
      }
      #pragma unroll
      for (int r = 0; r < 8; ++r)
        smem[cOff + (i * 16 + half * 8 + r) * STRIDE + j * 16 + m] = acc[r];
    }
  }
  wave_sync();
}

// smem[dOff..] = alpha*smem[sOff..] + beta*I   (in-place safe; lane l owns column l)
__device__ __forceinline__ void ew_axpb_diag(int dOff, int sOff,
                                             float alpha, float beta, int lane) {
  #pragma unroll
  for (int r = 0; r < 32; ++r) {
    float v = alpha * smem[sOff + r * STRIDE + lane];
    if (r == lane) v += beta;
    smem[dOff + r * STRIDE + lane] = v;
  }
  wave_sync();
}

// smem[dOff..] = beta * I
__device__ __forceinline__ void ew_ident(int dOff, float beta, int lane) {
  #pragma unroll
  for (int r = 0; r < 32; ++r)
    smem[dOff + r * STRIDE + lane] = (r == lane) ? beta : 0.f;
  wave_sync();
}

// Frobenius norm of 32x32 LDS matrix (wave32 xor-shuffle reduction)
__device__ __forceinline__ float frob(int sOff, int lane) {
  float s = 0.f;
  #pragma unroll
  for (int r = 0; r < 32; ++r) {
    float v = smem[sOff + r * STRIDE + lane];
    s += v * v;
  }
  #pragma unroll
  for (int off = 16; off > 0; off >>= 1)
    s += __shfl_xor(s, off, 32);
  return sqrtf(s);
}

// global (row-major contiguous 32x32) -> LDS, 16B vector loads, coalesced
__device__ __forceinline__ void load_mat(int dOff, const float* __restrict__ G, int lane) {
  #pragma unroll
  for (int t = 0; t < 8; ++t) {
    const int flat = t * 128 + lane * 4;     // col = (lane*4)&31 in {0..28}: never crosses a row
    const int r = flat >> 5, c = flat & 31;
    const float4 v = *reinterpret_cast<const float4*>(G + flat);   // global_load_b128
    const int p = dOff + r * STRIDE + c;
    smem[p] = v.x; smem[p + 1] = v.y; smem[p + 2] = v.z; smem[p + 3] = v.w;
  }
  wave_sync();
}

__device__ __forceinline__ void store_mat(float* __restrict__ G, int sOff, int lane) {
  #pragma unroll
  for (int t = 0; t < 8; ++t) {
    const int flat = t * 128 + lane * 4;
    const int r = flat >> 5, c = flat & 31;
    const int p = sOff + r * STRIDE + c;
    float4 v; v.x = smem[p]; v.y = smem[p + 1]; v.z = smem[p + 2]; v.w = smem[p + 3];
    *reinterpret_cast<float4*>(G + flat) = v;                      // global_store_b128
  }
}

// smem[dOff..] = smem[lOff..] + (0.5*(bias + bias^T) + logc*I)/64
// (exp argument, pre-scaled by 2^-6: the 2^6 log rescale cancels exp scaling)
__device__ __forceinline__ void ew_make_expo(int dOff, int lOff,
                                             const float* __restrict__ bias,
                                             float logc, int lane) {
  #pragma unroll
  for (int r = 0; r < 32; ++r) {
    float bs = 0.5f * (bias[r * 32 + lane] + bias[lane * 32 + r]);
    float v  = smem[lOff + r * STRIDE + lane] + 0.015625f * (bs + ((r == lane) ? logc : 0.f));
    smem[dOff + r * STRIDE + lane] = v;
  }
  wave_sync();
}

__global__ void __launch_bounds__(WAVES_PER_BLOCK * 32)
spd_logexp_kernel(const float* __restrict__ x, const float* __restrict__ bias,
                  float* __restrict__ out, int batch) {
  const int lane = threadIdx.x & 31;
  const int wave = threadIdx.x >> 5;
  const int mat  = blockIdx.x * WAVES_PER_BLOCK + wave;
  if (mat >= batch) return;                 // wave-uniform exit: EXEC all-ones for WMMA

  // per-wave LDS matrix buffers as integer offsets into smem[]
  int Y  = (wave * BUFS_PER_WAVE + 0) * MSZ;
  int Z  = (wave * BUFS_PER_WAVE + 1) * MSZ;
  int W1 = (wave * BUFS_PER_WAVE + 2) * MSZ;
  int W2 = (wave * BUFS_PER_WAVE + 3) * MSZ;

  const float* X = x + (size_t)mat * 1024;
  float* O       = out + (size_t)mat * 1024;
  __builtin_prefetch(X, 0, 0);              // global_prefetch_b8
  __builtin_prefetch(bias, 0, 3);

  // ---- load + spectral normalization: S = X/c, c = ||X||_F >= lambda_max ----
  load_mat(W2, X, lane);
  const float c    = fmaxf(frob(W2, lane), 1e-30f);
  const float logc = logf(c);
  ew_axpb_diag(Y, W2, 1.f / c, 0.f, lane);  // Y = S0, spectrum in (0, 1]

  // ---- inverse scaling-squaring: 6 matrix square roots via Denman-Beavers
  //      Newton-Schulz (inversion-free, 3 GEMMs/iter):
  //      T = 1.5I - 0.5*Z*Y ; Y <- Y*T ; Z <- T*Z ----
  const int ns_iters[6] = {14, 10, 7, 6, 5, 4};
  #pragma clang loop unroll(disable)
  for (int lvl = 0; lvl < 6; ++lvl) {
    ew_ident(Z, 1.f, lane);
    const int n = ns_iters[lvl];
    #pragma clang loop unroll(disable)
    for (int it = 0; it < n; ++it) {
      mm32(W1, Z, Y, lane);                     // W1 = Z*Y
      ew_axpb_diag(W1, W1, -0.5f, 1.5f, lane);  // W1 = T
      mm32(W2, Y, W1, lane);                    // W2 = newY
      mm32(Y,  W1, Z, lane);                    // old-Y buffer = newZ
      const int nY = W2, nZ = Y, s1 = Z, s2 = W1;
      Y = nY; Z = nZ; W1 = s1; W2 = s2;
    }
    // Y = sqrt of previous level; Z discarded (reset next level)
  }

  // ---- log(I+E) Taylor deg 8 (Horner), E = Y - I, ||E|| <~ 0.14 ----
  ew_axpb_diag(W1, Y, 1.f, -1.f, lane);     // W1 = E
  ew_ident(Z, -0.125f, lane);               // P = c8*I, c8 = -1/8
  int P = Z, Q = W2;
  #pragma clang loop unroll(disable)
  for (int k = 7; k >= 1; --k) {
    mm32(Q, W1, P, lane);                   // Q = E*P
    const float ck = ((k & 1) ? 1.f : -1.f) / (float)k;
    ew_axpb_diag(Q, Q, 1.f, ck, lane);      // Q += ck*I
    const int t = P; P = Q; Q = t;
  }
  mm32(Q, W1, P, lane);                     // Q = L = log(X^(1/64)) (un-rescaled)

  // ---- exp argument: Es = L + (bias_sym + logc*I)/64 ----
  ew_make_expo(Y, Q, bias, logc, lane);     // Y = Es (old Y dead)
  const int Ee = Y;
  int Pa = W1, Pb = P;                      // W1/P/Q now scratch

  // ---- exp(Es) Taylor deg 8 (Horner) ----
  const float invfact[8] = {1.f, 1.f, 0.5f, 1.f / 6.f, 1.f / 24.f,
                            1.f / 120.f, 1.f / 720.f, 1.f / 5040.f};
  ew_ident(Pa, 1.f / 40320.f, lane);        // P = I/8!
  #pragma clang loop unroll(disable)
  for (int k = 7; k >= 0; --k) {
    mm32(Pb, Ee, Pa, lane);                 // Pb = Es*P
    ew_axpb_diag(Pb, Pb, 1.f, invfact[k], lane);
    const int t = Pa; Pa = Pb; Pb = t;
  }
  // ---- 6 squarings: exp(A) = exp(Es)^(2^6) ----
  #pragma clang loop unroll(disable)
  for (int t = 0; t < 6; ++t) {
    mm32(Pb, Pa, Pa, lane);
    const int tmp = Pa; Pa = Pb; Pb = tmp;
  }

  store_mat(O, Pa, lane);
}

extern "C" void kernel_launch(void* const* d_in, const int* in_sizes, int n_in,
                              void* d_out, int out_size, void* d_ws, size_t ws_size,
                              hipStream_t stream) {
  const float* x    = (const float*)d_in[0];   // [B,32,32] f32
  const float* bias = (const float*)d_in[1];   // [32,32]   f32
  float* out        = (float*)d_out;           // [B,32,32] f32
  const int batch   = in_sizes[0] / 1024;

  // 132 KB dynamic LDS per block (8 waves x 4 matrix buffers, stride-33 pad);
  // WGP has 320 KB so 2 blocks (16 waves) co-resident per WGP.
  (void)hipFuncSetAttribute((const void*)spd_logexp_kernel,
                            hipFuncAttributeMaxDynamicSharedMemorySize,
                            (int)LDS_BYTES);
  const int grid = (batch + WAVES_PER_BLOCK - 1) / WAVES_PER_BLOCK;
  spd_logexp_kernel<<<grid, WAVES_PER_BLOCK * 32, LDS_BYTES, stream>>>(x, bias, out, batch);
}